// Grid_19971597926852
// MI455X (gfx1250) — compile-verified
//
#include <hip/hip_runtime.h>

typedef __attribute__((ext_vector_type(2))) float v2f;
typedef __attribute__((ext_vector_type(8))) float v8f;

constexpr int R  = 256, C = 256, NN = R * C;
constexpr int FN = 16, FE = 8;
constexpr int IN_DIM  = 2 * FN + FE + 5;   // 45
constexpr int OUT_DIM = 2 * FN + FE;       // 40
constexpr int KP   = 48;                   // K padded to multiple of 4
constexpr int KSTR = 50;                   // LDS row stride (bank spread, 8B aligned)
constexpr int E      = 586756;             // sum (256-|di|)(256-|dj|)
constexpr int ETILES = (E + 15) / 16;      // 36673

// Edge-block bases for offsets b=0..8, (di,dj) = (b/3-1, b%3-1)
__constant__ int c_base[9] = {0, 65025, 130305, 195330, 260610,
                              326146, 391426, 456451, 521731};

// ---------------------------------------------------------------- GEMM: out = inp @ W + b
// A (16 edges x 45 feat) from LDS, W transposed in LDS so each lane's B K-pair
// is one contiguous ds_load_b64. 36 v_wmma_f32_16x16x4_f32 per wave-tile.
__global__ __launch_bounds__(256)
void k_gemm(const float* __restrict__ data, const float* __restrict__ edata,
            const float* __restrict__ influx, const float* __restrict__ W,
            const float* __restrict__ bias, const float* __restrict__ rel,
            const int* __restrict__ src, const int* __restrict__ dst,
            float* __restrict__ out)
{
    __shared__ float sWt[48 * KSTR];     // W transposed: sWt[n*KSTR + k] (9.6 KB)
    __shared__ float sIn[8][16 * KSTR];  // per-wave 16x48 input tile (25.6 KB)

    const int tid  = threadIdx.x;
    const int wave = tid >> 5;
    const int lane = tid & 31;

    for (int i = tid; i < 48 * 48; i += 256) {
        int n = i / 48, k = i % 48;
        sWt[n * KSTR + k] = (k < IN_DIM && n < OUT_DIM) ? W[k * OUT_DIM + n] : 0.0f;
    }

    const int tile  = blockIdx.x * 8 + wave;
    const int ebase = tile * 16;
    float* my = sIn[wave];

    for (int idx = lane; idx < 16 * KP; idx += 32) {
        int m = idx / KP, k = idx % KP;
        int e  = ebase + m;
        int ee = (e < E) ? e : (E - 1);
        float v;
        if (k < FN)                  v = data[src[ee] * FN + k];
        else if (k < 2 * FN)         v = data[dst[ee] * FN + (k - FN)];
        else if (k < 2 * FN + FE)    v = edata[(size_t)ee * FE + (k - 2 * FN)];
        else if (k < 2 * FN + FE + 3)v = rel[(size_t)ee * 3 + (k - 2 * FN - FE)];
        else if (k == 43)            v = influx[src[ee]];
        else if (k == 44)            v = influx[dst[ee]];
        else                         v = 0.0f;
        my[m * KSTR + k] = v;
    }
    __syncthreads();

    const int half = lane >> 4;   // K-pair selector for A/B, row-half for D
    const int sub  = lane & 15;   // M for A, N for B/D

    v8f acc[3];
    #pragma unroll
    for (int t = 0; t < 3; ++t) {
        int n = t * 16 + sub;
        float bv = (n < OUT_DIM) ? bias[n] : 0.0f;
        #pragma unroll
        for (int r2 = 0; r2 < 8; ++r2) acc[t][r2] = bv;
    }

    #pragma unroll
    for (int kc = 0; kc < KP / 4; ++kc) {
        const int kb = kc * 4 + 2 * half;
        v2f a;
        a[0] = my[sub * KSTR + kb];
        a[1] = my[sub * KSTR + kb + 1];
        #pragma unroll
        for (int t = 0; t < 3; ++t) {
            v2f b;
            b[0] = sWt[(t * 16 + sub) * KSTR + kb];
            b[1] = sWt[(t * 16 + sub) * KSTR + kb + 1];
            acc[t] = __builtin_amdgcn_wmma_f32_16x16x4_f32(
                false, a, false, b, (short)0, acc[t], false, false);
        }
    }

    #pragma unroll
    for (int t = 0; t < 3; ++t) {
        #pragma unroll
        for (int r2 = 0; r2 < 8; ++r2) {
            int e = ebase + r2 + 8 * half;
            int n = t * 16 + sub;
            if (e < E && n < OUT_DIM) out[(size_t)e * OUT_DIM + n] = acc[t][r2];
        }
    }
}

// ------------------------------------------------ per-node aggregation + src-softmax
__global__ __launch_bounds__(256)
void k_node(const float* __restrict__ out, float* __restrict__ data_out,
            float* __restrict__ edata_out)
{
    int n = blockIdx.x * 256 + threadIdx.x;
    if (n >= NN) return;
    const int r = n >> 8, c = n & (C - 1);

    float acc[FN];
    #pragma unroll
    for (int k = 0; k < FN; ++k) acc[k] = 0.0f;
    float wsum = 0.0f;
    float flux[9];
    int   oe[9];

    #pragma unroll
    for (int b = 0; b < 9; ++b) {
        const int di = b / 3 - 1, dj = b % 3 - 1;
        const int r0 = (di < 0) ? 1 : 0, c0 = (dj < 0) ? 1 : 0;
        const int w  = C - ((dj == 0) ? 0 : 1);

        // out-edge: (r,c) -> (r+di, c+dj)
        int e = -1;
        if ((unsigned)(r + di) < (unsigned)R && (unsigned)(c + dj) < (unsigned)C)
            e = c_base[b] + (r - r0) * w + (c - c0);
        oe[b] = e; flux[b] = 0.0f;
        if (e >= 0) {
            const float* o = out + (size_t)e * OUT_DIM;
            float wa = fmaxf(o[0], 0.0f);
            #pragma unroll
            for (int k = 0; k < FN; ++k) acc[k] += o[k] * wa;
            wsum += wa;
            flux[b] = fmaxf(o[2 * FN], 0.0f);
            #pragma unroll
            for (int j = 1; j < FE; ++j)
                edata_out[(size_t)e * FE + j] = o[2 * FN + j];
        }
        // in-edge: (r-di, c-dj) -> (r,c)
        int rs = r - di, cs = c - dj;
        if ((unsigned)rs < (unsigned)R && (unsigned)cs < (unsigned)C) {
            int e2 = c_base[b] + (rs - r0) * w + (cs - c0);
            const float* o = out + (size_t)e2 * OUT_DIM + FN;
            float wb = fmaxf(o[0], 0.0f);
            #pragma unroll
            for (int k = 0; k < FN; ++k) acc[k] += o[k] * wb;
            wsum += wb;
        }
    }
    const float inv = 1.0f / fmaxf(wsum, 1e-6f);
    #pragma unroll
    for (int k = 0; k < FN; ++k) data_out[(size_t)n * FN + k] = acc[k] * inv;

    // softmax over this node's out-edges (every node has the self-edge, flux >= 0)
    float mx = 0.0f;
    #pragma unroll
    for (int b = 0; b < 9; ++b) if (oe[b] >= 0) mx = fmaxf(mx, flux[b]);
    float ex[9]; float se = 0.0f;
    #pragma unroll
    for (int b = 0; b < 9; ++b) { ex[b] = (oe[b] >= 0) ? expf(flux[b] - mx) : 0.0f; se += ex[b]; }
    const float rinv = 1.0f / se;
    #pragma unroll
    for (int b = 0; b < 9; ++b)
        if (oe[b] >= 0) edata_out[(size_t)oe[b] * FE] = ex[b] * rinv;
}

// ------------------------------------------------ transport (gather form over in-edges)
__global__ __launch_bounds__(256)
void k_transport(const float* __restrict__ edata, const float* __restrict__ mat_in,
                 const float* __restrict__ fuel_in, const float* __restrict__ influx,
                 float* __restrict__ mat_out, float* __restrict__ fuel_out,
                 float* __restrict__ pc, float* __restrict__ pf)
{
    __shared__ float sc[256], sf[256];
    const int tid = threadIdx.x;
    const int n = blockIdx.x * 256 + tid;
    float nm = 0.0f, nf = 0.0f, cons = 0.0f;
    if (n < NN) {
        const int r = n >> 8, c = n & (C - 1);
        #pragma unroll
        for (int b = 0; b < 9; ++b) {
            const int di = b / 3 - 1, dj = b % 3 - 1;
            const int r0 = (di < 0) ? 1 : 0, c0 = (dj < 0) ? 1 : 0;
            const int w  = C - ((dj == 0) ? 0 : 1);
            const float len = (di * di + dj * dj == 2) ? 1.41421356237309515f
                            : (di * di + dj * dj == 1) ? 1.0f : 0.0f;
            int rs = r - di, cs = c - dj;
            if ((unsigned)rs < (unsigned)R && (unsigned)cs < (unsigned)C) {
                int e = c_base[b] + (rs - r0) * w + (cs - c0);
                int s = (rs << 8) | cs;
                float m  = mat_in[s] + fmaxf(influx[s], 0.0f);
                float tm = edata[(size_t)e * FE] * m;
                nm += tm;
                nf += tm * (len + fuel_in[s]);
            }
        }
        float ineg = fmaxf(-influx[n], 0.0f);
        cons = fminf(nm, ineg);
        mat_out[n]  = nm - cons;
        fuel_out[n] = nf;
    }
    sc[tid] = cons; sf[tid] = nf;
    __syncthreads();
    for (int s = 128; s > 0; s >>= 1) {
        if (tid < s) { sc[tid] += sc[tid + s]; sf[tid] += sf[tid + s]; }
        __syncthreads();
    }
    if (tid == 0) { pc[blockIdx.x] = sc[0]; pf[blockIdx.x] = sf[0]; }
}

// ------------------------------------------------ small utility kernels
__global__ void k_init(float* __restrict__ matA, float* __restrict__ fuelA,
                       float* __restrict__ scal)
{
    int i = blockIdx.x * 256 + threadIdx.x;
    if (i < NN) { matA[i] = 0.0f; fuelA[i] = 0.0f; }
    if (i < 4) scal[i] = 0.0f;
}

__global__ void k_sum_part(const float* __restrict__ x, float* __restrict__ part)
{
    __shared__ float s[256];
    int tid = threadIdx.x;
    s[tid] = x[blockIdx.x * 256 + tid];            // NN == 256*256 exactly
    __syncthreads();
    for (int k = 128; k > 0; k >>= 1) {
        if (tid < k) s[tid] += s[tid + k];
        __syncthreads();
    }
    if (tid == 0) part[blockIdx.x] = s[0];
}

__global__ void k_final_sum(const float* __restrict__ part, float* __restrict__ scal)
{
    __shared__ float s[256];
    int tid = threadIdx.x;
    s[tid] = part[tid];
    __syncthreads();
    for (int k = 128; k > 0; k >>= 1) {
        if (tid < k) s[tid] += s[tid + k];
        __syncthreads();
    }
    if (tid == 0) scal[0] = s[0];
}

__global__ void k_final_acc2(const float* __restrict__ pc, const float* __restrict__ pf,
                             float* __restrict__ scal)
{
    __shared__ float s1[256], s2[256];
    int tid = threadIdx.x;
    s1[tid] = pc[tid]; s2[tid] = pf[tid];
    __syncthreads();
    for (int k = 128; k > 0; k >>= 1) {
        if (tid < k) { s1[tid] += s1[tid + k]; s2[tid] += s2[tid + k]; }
        __syncthreads();
    }
    if (tid == 0) { scal[1] += s1[0]; scal[2] += s2[0]; }
}

__global__ void k_influx(const float* __restrict__ raw, const float* __restrict__ scal,
                         float* __restrict__ influx)
{
    int i = blockIdx.x * 256 + threadIdx.x;
    if (i < NN) influx[i] = raw[i] - scal[0] * (1.0f / NN);
}

__global__ void k_out(const float* __restrict__ data, const float* __restrict__ mat,
                      const float* __restrict__ scal, float* __restrict__ o)
{
    int i = blockIdx.x * 256 + threadIdx.x;
    const int D = NN * FN;
    if (i < D)                 o[i] = data[i];
    else if (i < D + NN)       o[i] = mat[i - D];
    else if (i == D + NN)      o[i] = scal[1];   // tot_cons
    else if (i == D + NN + 1)  o[i] = scal[2];   // tot_fuel
}

// ---------------------------------------------------------------- launcher
extern "C" void kernel_launch(void* const* d_in, const int* in_sizes, int n_in,
                              void* d_out, int out_size, void* d_ws, size_t ws_size,
                              hipStream_t stream)
{
    const float* node_data  = (const float*)d_in[0];
    const float* edge_data  = (const float*)d_in[1];
    const float* influx_raw = (const float*)d_in[2];
    const float* W          = (const float*)d_in[3];
    const float* bvec       = (const float*)d_in[4];
    const float* rel        = (const float*)d_in[5];
    const int*   src        = (const int*)d_in[6];
    const int*   dst        = (const int*)d_in[7];
    float* o = (float*)d_out;

    float* ws = (float*)d_ws;
    size_t off = 0;
    float* outbuf = ws + off; off += (size_t)E * OUT_DIM;   // 93.9 MB
    float* dataA  = ws + off; off += (size_t)NN * FN;
    float* dataB  = ws + off; off += (size_t)NN * FN;
    float* edataA = ws + off; off += (size_t)E * FE;
    float* edataB = ws + off; off += (size_t)E * FE;
    float* influx = ws + off; off += NN;
    float* matA   = ws + off; off += NN;
    float* matB   = ws + off; off += NN;
    float* fuelA  = ws + off; off += NN;
    float* fuelB  = ws + off; off += NN;
    float* pc     = ws + off; off += 256;
    float* pf     = ws + off; off += 256;
    float* scal   = ws + off; off += 4;   // [0]=sum(influx_raw) [1]=tot_cons [2]=tot_fuel

    hipMemcpyAsync(dataA, node_data, sizeof(float) * (size_t)NN * FN,
                   hipMemcpyDeviceToDevice, stream);
    hipMemcpyAsync(edataA, edge_data, sizeof(float) * (size_t)E * FE,
                   hipMemcpyDeviceToDevice, stream);

    k_init<<<NN / 256, 256, 0, stream>>>(matA, fuelA, scal);
    k_sum_part<<<256, 256, 0, stream>>>(influx_raw, pc);
    k_final_sum<<<1, 256, 0, stream>>>(pc, scal);
    k_influx<<<NN / 256, 256, 0, stream>>>(influx_raw, scal, influx);

    const int gemmBlocks = (ETILES + 7) / 8;
    float *dIn = dataA, *dOut = dataB, *eIn = edataA, *eOut = edataB;
    for (int it = 0; it < 3; ++it) {
        k_gemm<<<gemmBlocks, 256, 0, stream>>>(dIn, eIn, influx, W, bvec, rel,
                                               src, dst, outbuf);
        k_node<<<NN / 256, 256, 0, stream>>>(outbuf, dOut, eOut);
        float* t;
        t = dIn; dIn = dOut; dOut = t;
        t = eIn; eIn = eOut; eOut = t;
    }

    float *mIn = matA, *mOut = matB, *fIn = fuelA, *fOut = fuelB;
    for (int it = 0; it < 3; ++it) {
        k_transport<<<NN / 256, 256, 0, stream>>>(eIn, mIn, fIn, influx,
                                                  mOut, fOut, pc, pf);
        k_final_acc2<<<1, 256, 0, stream>>>(pc, pf, scal);
        float* t;
        t = mIn; mIn = mOut; mOut = t;
        t = fIn; fIn = fOut; fOut = t;
    }

    const int TOTAL = NN * FN + NN + 2;
    k_out<<<(TOTAL + 255) / 256, 256, 0, stream>>>(dIn, mIn, scal, o);
}